// Net_34136400069284
// MI455X (gfx1250) — compile-verified
//
#include <hip/hip_runtime.h>
#include <math.h>

// ---------------------------------------------------------------------------
// MI455X / gfx1250 fused implementation:
//   conv(VALID) -> per-pixel 257th-order-statistic -> sigmoid -> fused pooling
// GEMM view: M = 128*729 = 93312, N = 1024, K = 108 (20.6 GFLOP, fp32 WMMA).
// Whole working set < 192MB L2 -> fuse to avoid the 382MB conv intermediate.
// A-tile / bias staging uses GLOBAL_LOAD_ASYNC_TO_LDS (ASYNCcnt) — CDNA5 path.
// ---------------------------------------------------------------------------

typedef __attribute__((ext_vector_type(2))) float v2f;
typedef __attribute__((ext_vector_type(8))) float v8f;

#define BATCH   128
#define CIN     3
#define IN_HW   32
#define COUT    1024
#define CPAD    1040          // +16 floats: kills LDS bank conflicts on C stores
#define KDIM    108           // 3*6*6
#define KPAD    112
#define OHW     27
#define NPIX    (OHW*OHW)     // 729
#define MTOT    (BATCH*NPIX)  // 93312 (divisible by 16)
#define TILE_M  16
#define OUTSZ   (BATCH*COUT*16)   // one branch: (128,1024,4,4)

__global__ void zero_out_k(float* __restrict__ out, int n) {
  int i = blockIdx.x * blockDim.x + threadIdx.x;
  if (i < n) out[i] = 0.0f;
}

// kernel (1024,3,6,6) row-major -> K-major wT[k*1024 + n]; 432KB, stays in L2.
__global__ void transpose_w_k(const float* __restrict__ w, float* __restrict__ wT) {
  int i = blockIdx.x * blockDim.x + threadIdx.x;
  if (i < COUT * KDIM) {
    int n = i / KDIM, k = i - n * KDIM;
    wT[k * COUT + n] = w[i];
  }
}

// monotone float <-> orderable uint (total order incl. negatives)
__device__ __forceinline__ unsigned f2key(float f) {
  unsigned u = __float_as_uint(f);
  return (u & 0x80000000u) ? ~u : (u | 0x80000000u);
}
__device__ __forceinline__ float key2f(unsigned k) {
  return __uint_as_float((k & 0x80000000u) ? (k ^ 0x80000000u) : ~k);
}

// CDNA5 async copy: memory -> LDS without touching VGPR return paths.
// LDS byte address = low 32 bits of the generic pointer (ISA aperture rule).
// Tracked by ASYNCcnt.
__device__ __forceinline__ void async_g2lds_b32(const float* gptr, float* lds_ptr) {
  unsigned lds_off = (unsigned)(size_t)lds_ptr;
  unsigned long long ga = (unsigned long long)(size_t)gptr;
  asm volatile("global_load_async_to_lds_b32 %0, %1, off"
               :: "v"(lds_off), "v"(ga) : "memory");
}
__device__ __forceinline__ void async_wait_all() {
  asm volatile("s_wait_asynccnt 0x0" ::: "memory");
}

__launch_bounds__(256, 1)
__global__ void fused_k(const float* __restrict__ x, const float* __restrict__ wT,
                        const float* __restrict__ bias, float* __restrict__ out)
{
  // 76KB LDS -> 4 workgroups per 320KB WGP, 8 wave32 each.
  __shared__ float sA[TILE_M][KPAD];     // im2col tile, 7KB
  __shared__ float sC[TILE_M][CPAD];     // conv results, 65KB (padded stride)
  __shared__ float sBias[COUT];          // 4KB

  const int tid  = threadIdx.x;
  const int lane = tid & 31;
  const int wave = tid >> 5;
  const int pix0 = blockIdx.x * TILE_M;

  // ---- stage im2col A tile (16 rows x 108) + bias: async -> LDS ----
  for (int i = tid; i < TILE_M * KDIM; i += 256) {
    int row = i / KDIM, k = i - row * KDIM;
    int p = pix0 + row;
    int b = p / NPIX, rem = p - b * NPIX;
    int oh = rem / OHW, ow = rem - oh * OHW;
    int ci = k / 36, r = (k - ci * 36) / 6, c = k % 6;
    async_g2lds_b32(&x[((b * CIN + ci) * IN_HW + (oh + r)) * IN_HW + (ow + c)],
                    &sA[row][k]);
  }
  for (int i = tid; i < COUT; i += 256)
    async_g2lds_b32(&bias[i], &sBias[i]);
  async_wait_all();
  __syncthreads();

  // ---- fp32 WMMA GEMM: each wave owns 128 channels = 8 N-tiles of 16 ----
  v8f acc[8] = {};
  // A 16x4 layout (ISA 7.12.2): lanes 0-15 -> VGPR0:K=0, VGPR1:K=1;
  //                             lanes 16-31 -> VGPR0:K=2, VGPR1:K=3.
  const int arow = lane & 15;
  const int koff = (lane >> 4) << 1;               // 0 or 2
  const int ncol = wave * 128 + (lane & 15);       // B column per lane

  for (int k0 = 0; k0 < KDIM; k0 += 4) {
    v2f av;
    av.x = sA[arow][k0 + koff];
    av.y = sA[arow][k0 + koff + 1];
    const float* b0 = wT + (size_t)(k0 + koff) * COUT + ncol;
    const float* b1 = b0 + COUT;
    if (k0 + 4 < KDIM) __builtin_prefetch(b0 + 4 * COUT, 0, 3);  // near-cache prefetch
#pragma unroll
    for (int t = 0; t < 8; ++t) {
      v2f bv;                      // B 4x16 fragment, mirrored K assignment
      bv.x = b0[t * 16];
      bv.y = b1[t * 16];
      acc[t] = __builtin_amdgcn_wmma_f32_16x16x4_f32(
          false, av, false, bv, (short)0, acc[t], false, false);
    }
  }

  // ---- C tiles -> LDS [pixel][channel] (C layout: VGPR j => M=j or j+8) ----
  {
    const int n0 = wave * 128 + (lane & 15);
    const int mb = (lane >> 4) * 8;
#pragma unroll
    for (int t = 0; t < 8; ++t)
#pragma unroll
      for (int j = 0; j < 8; ++j)
        sC[mb + j][n0 + t * 16] = acc[t][j];
  }
  __syncthreads();

  // ---- per-pixel epilogue: 2 pixels per wave ----
  float bb[32];
#pragma unroll
  for (int i = 0; i < 32; ++i) bb[i] = sBias[i * 32 + lane];

  for (int mp = 0; mp < 2; ++mp) {
    const int m = wave * 2 + mp;
    const int p = pix0 + m;
    const int b = p / NPIX, rem = p - b * NPIX;
    const int oh = rem / OHW, ow = rem - oh * OHW;

    // fused avgpool(6,3) + 2x2 blockmean: final cell (i,j) covers rows 6i..6i+8
    // with separable weights {1,1,1,2,2,2,1,1,1}/12 per dimension.
    int ri[2], rj[2]; float rw[2], cw[2];
    int nr = 0, nc = 0;
    for (int ii = 0; ii < 4; ++ii) {
      int d = oh - 6 * ii;
      if (d >= 0 && d <= 8) { ri[nr] = ii; rw[nr] = (d >= 3 && d <= 5) ? (2.f/12.f) : (1.f/12.f); ++nr; }
      d = ow - 6 * ii;
      if (d >= 0 && d <= 8) { rj[nc] = ii; cw[nc] = (d >= 3 && d <= 5) ? (2.f/12.f) : (1.f/12.f); ++nc; }
    }

    float vals[32];
#pragma unroll
    for (int i = 0; i < 32; ++i) vals[i] = sC[m][i * 32 + lane];  // conflict-free

    for (int s = 0; s < 2; ++s) {      // s=0: bias-out (out1), s=1: bias+out (out2)
      unsigned keys[32];
#pragma unroll
      for (int i = 0; i < 32; ++i) {
        float t = s ? (bb[i] + vals[i]) : (bb[i] - vals[i]);
        keys[i] = f2key(t);
      }
      // exact 257th smallest (0-based index 256): MSB-first radix select,
      // register-resident, wave32 reduction via __shfl_xor.
      unsigned prefix = 0;
      int remaining = 256;
      for (int bit = 31; bit >= 0; --bit) {
        int cl = 0;
#pragma unroll
        for (int i = 0; i < 32; ++i) {
          unsigned d = keys[i] ^ prefix;
          cl += (int)((((d >> bit) >> 1) == 0u) & (((keys[i] >> bit) & 1u) == 0u));
        }
#pragma unroll
        for (int off = 16; off > 0; off >>= 1) cl += __shfl_xor(cl, off, 32);
        if (remaining >= cl) { remaining -= cl; prefix |= (1u << bit); }
      }
      const float thr = key2f(prefix);

      const long obase = (long)s * OUTSZ + (long)b * COUT * 16;
#pragma unroll
      for (int i = 0; i < 32; ++i) {
        float t   = key2f(keys[i]);
        float msk = 1.0f / (1.0f + __expf(t - thr));   // sigmoid(thr - t)
        long cbase = obase + (long)(i * 32 + lane) * 16;
        for (int a = 0; a < nr; ++a)
          for (int e = 0; e < nc; ++e)
            atomicAdd(&out[cbase + ri[a] * 4 + rj[e]], rw[a] * cw[e] * msk);
      }
    }
  }
}

extern "C" void kernel_launch(void* const* d_in, const int* in_sizes, int n_in,
                              void* d_out, int out_size, void* d_ws, size_t ws_size,
                              hipStream_t stream) {
  (void)in_sizes; (void)n_in; (void)out_size; (void)ws_size;
  const float* x    = (const float*)d_in[0];
  const float* w    = (const float*)d_in[1];
  const float* bias = (const float*)d_in[2];
  float* out = (float*)d_out;
  float* wT  = (float*)d_ws;                    // 110592 floats = 432KB scratch

  zero_out_k<<<(2 * OUTSZ + 255) / 256, 256, 0, stream>>>(out, 2 * OUTSZ);
  transpose_w_k<<<(COUT * KDIM + 255) / 256, 256, 0, stream>>>(w, wT);
  fused_k<<<MTOT / TILE_M, 256, 0, stream>>>(x, wT, bias, out);
}